// NIC_Decoder_52673478918507
// MI455X (gfx1250) — compile-verified
//
#include <hip/hip_runtime.h>
#include <math.h>

// ---------------------------------------------------------------------------
// NIC decoder (LSTM caption head) for MI455X / gfx1250, fp32 via
// V_WMMA_F32_16X16X4_F32 (exact fp32 matrix math, wave32).
//
// B=64, T=33, E=U=512, V=10000, 4U=2048, M=B*T=2112
//
// Round-2 change: register-level M-blocking (MB=2) in both big GEMMs so each
// B fragment feeds two WMMAs -> ~2x arithmetic intensity vs L2 traffic.
// ---------------------------------------------------------------------------

typedef __attribute__((ext_vector_type(2))) float v2f;
typedef __attribute__((ext_vector_type(8))) float v8f;

static constexpr int Bn = 64;
static constexpr int Tn = 33;
static constexpr int En = 512;
static constexpr int Un = 512;
static constexpr int Vn = 10000;
static constexpr int G4 = 2048;            // 4*U
static constexpr int Mn = Bn * Tn;         // 2112

__device__ __forceinline__ v8f wmma4(v2f a, v2f b, v8f c) {
  // 8 args: (neg_a, A, neg_b, B, c_mod, C, reuse_a, reuse_b)
  return __builtin_amdgcn_wmma_f32_16x16x4_f32(false, a, false, b,
                                               (short)0, c, false, false);
}

__device__ __forceinline__ float sigmoidf_(float x) {
  return 1.0f / (1.0f + __expf(-x));
}

// ---------------------------------------------------------------------------
// Kernel 1: build x = [images_emb ; emb_table[targets[:, :-1]] * word_mask]
// x is (B, T, E) row-major, flattened rows m = b*T + t.
// ---------------------------------------------------------------------------
__global__ void build_x_kernel(const float* __restrict__ img,
                               const float* __restrict__ emb,
                               const int* __restrict__ tgt,
                               const float* __restrict__ wmask,
                               float* __restrict__ x) {
  int idx = blockIdx.x * blockDim.x + threadIdx.x;   // over B*T*E (exact grid)
  int e = idx % En;
  int m = idx / En;
  int t = m % Tn;
  int b = m / Tn;
  float v;
  if (t == 0) {
    v = img[b * En + e];
  } else {
    int w = tgt[b * Tn + (t - 1)];
    v = emb[w * En + e] * wmask[(t - 1) * En + e];
  }
  x[idx] = v;
}

__global__ void zero_kernel(float* __restrict__ p, int n) {
  int i = blockIdx.x * blockDim.x + threadIdx.x;
  if (i < n) p[i] = 0.0f;
}

// ---------------------------------------------------------------------------
// Kernel 2: z_all = x @ Wx + b     (M=2112, K=512, N=2048)
// One wave computes a 32 x 64 strip: MB=2 A fragments x NB=4 B fragments,
// 8 WMMAs per K-step, each B fragment reused for both M tiles.
// ---------------------------------------------------------------------------
__global__ void gemm_zall_kernel(const float* __restrict__ A,     // (M, E)
                                 const float* __restrict__ Wx,    // (E, 4U)
                                 const float* __restrict__ bias,  // (4U)
                                 float* __restrict__ C) {         // (M, 4U)
  constexpr int NB = 4;
  constexpr int MB = 2;
  constexpr int NGROUPS = (G4 / 16) / NB;  // 32
  int wave = blockIdx.x * (blockDim.x >> 5) + (threadIdx.x >> 5);
  int mg = wave / NGROUPS;                 // 0..65
  int ng = wave % NGROUPS;
  if (mg >= Mn / (16 * MB)) return;        // uniform per-wave; grid is exact
  int lane = threadIdx.x & 31;
  int half = lane >> 4;
  int l = lane & 15;
  int arow0 = mg * 32 + l;                 // A: M = lane&15 in both halves
  int arow1 = arow0 + 16;

  v8f acc[MB][NB] = {};
#pragma unroll 2
  for (int kk = 0; kk < En; kk += 4) {
    int k2 = kk + half * 2;                // lane half -> K block {0,1}/{2,3}
    v2f a0 = *(const v2f*)(A + arow0 * En + k2);
    v2f a1 = *(const v2f*)(A + arow1 * En + k2);
#pragma unroll
    for (int j = 0; j < NB; ++j) {
      int col = (ng * NB + j) * 16 + l;
      v2f bf;
      bf.x = Wx[k2 * G4 + col];
      bf.y = Wx[(k2 + 1) * G4 + col];
      acc[0][j] = wmma4(a0, bf, acc[0][j]);
      acc[1][j] = wmma4(a1, bf, acc[1][j]);
    }
  }
#pragma unroll
  for (int j = 0; j < NB; ++j) {
    int col = (ng * NB + j) * 16 + l;
    float bv = bias[col];
#pragma unroll
    for (int i = 0; i < MB; ++i) {
#pragma unroll
      for (int r = 0; r < 8; ++r) {
        int row = mg * 32 + i * 16 + r + 8 * half;  // C/D row layout
        C[row * G4 + col] = acc[i][j][r] + bv;
      }
    }
  }
}

// ---------------------------------------------------------------------------
// Kernel 3: one LSTM step. Each wave owns a (16 batch x 16 hidden) slice and
// computes all 4 gate GEMMs (recurrent-dropout-scaled A, shared h fragment)
// plus the elementwise c/h update. 128 waves per step (latency-bound).
// ---------------------------------------------------------------------------
__global__ void lstm_step_kernel(const float* __restrict__ z_all,   // (B,T,4U)
                                 int t,
                                 const float* __restrict__ Uh,      // (U,4U)
                                 const float* __restrict__ rmask,   // (B,4,U)
                                 const float* __restrict__ h_in,    // (B,U)
                                 float* __restrict__ h_out,         // (B,U)
                                 float* __restrict__ cbuf,          // (B,U)
                                 float* __restrict__ hs) {          // (B,T,U)
  int wave = blockIdx.x * (blockDim.x >> 5) + (threadIdx.x >> 5);
  int mtile = wave >> 5;                   // 4 M tiles (B=64)
  int ntile = wave & 31;                   // 32 N tiles (U=512)
  int lane = threadIdx.x & 31;
  int half = lane >> 4;
  int l = lane & 15;
  int hrow = mtile * 16 + l;

  v8f acc[4] = {};
#pragma unroll 2
  for (int kk = 0; kk < Un; kk += 4) {
    int k2 = kk + half * 2;
    v2f hf = *(const v2f*)(h_in + hrow * Un + k2);
#pragma unroll
    for (int g = 0; g < 4; ++g) {
      v2f rm = *(const v2f*)(rmask + (hrow * 4 + g) * Un + k2);
      v2f a;
      a.x = hf.x * rm.x;                   // hm = h * rec_mask[:,g,:]
      a.y = hf.y * rm.y;
      int col = g * Un + ntile * 16 + l;
      v2f bf;
      bf.x = Uh[k2 * G4 + col];
      bf.y = Uh[(k2 + 1) * G4 + col];
      acc[g] = wmma4(a, bf, acc[g]);
    }
  }

#pragma unroll
  for (int r = 0; r < 8; ++r) {
    int brow = mtile * 16 + r + 8 * half;
    int vcol = ntile * 16 + l;
    const float* zrow = z_all + (brow * Tn + t) * G4;
    float zi = zrow[0 * Un + vcol] + acc[0][r];
    float zf = zrow[1 * Un + vcol] + acc[1][r];
    float zg = zrow[2 * Un + vcol] + acc[2][r];
    float zo = zrow[3 * Un + vcol] + acc[3][r];
    float ig = sigmoidf_(zi);
    float fg = sigmoidf_(zf);
    float gg = tanhf(zg);
    float og = sigmoidf_(zo);
    float cn = fg * cbuf[brow * Un + vcol] + ig * gg;
    float hn = og * tanhf(cn);
    cbuf[brow * Un + vcol] = cn;           // c owned exclusively by this wave
    h_out[brow * Un + vcol] = hn;          // ping-pong buffer (h_in != h_out)
    hs[(brow * Tn + t) * Un + vcol] = hn;
  }
}

// ---------------------------------------------------------------------------
// Kernel 4: logits = (hs * final_mask) @ Wd + bd   (M=2112, K=512, N=10000)
// One wave computes 32 x 80: MB=2 x NB=5 -> 10 WMMAs per K-step,
// every Wd fragment reused for two M tiles (~11 FLOP/byte of L2 traffic).
// ---------------------------------------------------------------------------
__global__ void gemm_head_kernel(const float* __restrict__ hs,     // (M,U)
                                 const float* __restrict__ fmask,  // (T,U)
                                 const float* __restrict__ Wd,     // (U,V)
                                 const float* __restrict__ bd,     // (V)
                                 float* __restrict__ out) {        // (M,V)
  constexpr int NB = 5;
  constexpr int MB = 2;
  constexpr int NGROUPS = (Vn / 16) / NB;  // 125
  int wave = blockIdx.x * (blockDim.x >> 5) + (threadIdx.x >> 5);
  int mg = wave / NGROUPS;                 // 0..65
  int ng = wave % NGROUPS;
  if (mg >= Mn / (16 * MB)) return;
  int lane = threadIdx.x & 31;
  int half = lane >> 4;
  int l = lane & 15;
  int arow0 = mg * 32 + l;
  int arow1 = arow0 + 16;
  int at0 = arow0 % Tn;                    // final_mask broadcast over batch
  int at1 = arow1 % Tn;

  v8f acc[MB][NB] = {};
#pragma unroll 2
  for (int kk = 0; kk < Un; kk += 4) {
    int k2 = kk + half * 2;
    v2f h0 = *(const v2f*)(hs + arow0 * Un + k2);
    v2f m0 = *(const v2f*)(fmask + at0 * Un + k2);
    v2f h1 = *(const v2f*)(hs + arow1 * Un + k2);
    v2f m1 = *(const v2f*)(fmask + at1 * Un + k2);
    v2f a0, a1;
    a0.x = h0.x * m0.x;
    a0.y = h0.y * m0.y;
    a1.x = h1.x * m1.x;
    a1.y = h1.y * m1.y;
#pragma unroll
    for (int j = 0; j < NB; ++j) {
      int col = (ng * NB + j) * 16 + l;
      v2f bf;
      bf.x = Wd[k2 * Vn + col];
      bf.y = Wd[(k2 + 1) * Vn + col];
      acc[0][j] = wmma4(a0, bf, acc[0][j]);
      acc[1][j] = wmma4(a1, bf, acc[1][j]);
    }
  }
#pragma unroll
  for (int j = 0; j < NB; ++j) {
    int col = (ng * NB + j) * 16 + l;
    float bv = bd[col];
#pragma unroll
    for (int i = 0; i < MB; ++i) {
#pragma unroll
      for (int r = 0; r < 8; ++r) {
        int row = mg * 32 + i * 16 + r + 8 * half;
        out[row * Vn + col] = acc[i][j][r] + bv;
      }
    }
  }
}

// ---------------------------------------------------------------------------
// Launch. Inputs (setup_inputs order):
// 0 images_emb (B,E) 1 emb_table (V,E) 2 Wx (E,4U) 3 Uh (U,4U) 4 b (4U)
// 5 Wd (U,V) 6 bd (V) 7 word_mask (1,T-1,E) 8 rec_mask (B,4,U)
// 9 final_mask (1,T,U) 10 targets (B,T) int
// ---------------------------------------------------------------------------
extern "C" void kernel_launch(void* const* d_in, const int* in_sizes, int n_in,
                              void* d_out, int out_size, void* d_ws,
                              size_t ws_size, hipStream_t stream) {
  const float* img   = (const float*)d_in[0];
  const float* emb   = (const float*)d_in[1];
  const float* Wx    = (const float*)d_in[2];
  const float* Uh    = (const float*)d_in[3];
  const float* bb    = (const float*)d_in[4];
  const float* Wd    = (const float*)d_in[5];
  const float* bd    = (const float*)d_in[6];
  const float* wmask = (const float*)d_in[7];
  const float* rmask = (const float*)d_in[8];
  const float* fmask = (const float*)d_in[9];
  const int*   tgt   = (const int*)d_in[10];
  float* out = (float*)d_out;

  // Workspace layout (floats): ~26.3 MB total
  float* ws    = (float*)d_ws;
  float* x     = ws;                       // M*E      = 1,081,344
  float* zall  = x + Mn * En;              // M*4U     = 4,325,376
  float* hsbuf = zall + Mn * G4;           // M*U      = 1,081,344
  float* h0    = hsbuf + Mn * Un;          // B*U      = 32,768
  float* h1    = h0 + Bn * Un;             // B*U
  float* cbuf  = h1 + Bn * Un;             // B*U

  // 1) build x (exact grid: 2112*512 / 256 = 4224 blocks)
  build_x_kernel<<<(Mn * En) / 256, 256, 0, stream>>>(img, emb, tgt, wmask, x);

  // 2) zero h0, h1, c (contiguous: 3 * 32768 = 98304; re-init every call)
  zero_kernel<<<(3 * Bn * Un + 255) / 256, 256, 0, stream>>>(h0, 3 * Bn * Un);

  // 3) z_all = x @ Wx + b : 66 m-groups * 32 n-groups = 2112 waves, 4/block
  gemm_zall_kernel<<<2112 / 4, 128, 0, stream>>>(x, Wx, bb, zall);

  // 4) LSTM scan: 33 sequential step kernels (stream order = dependency)
  for (int t = 0; t < Tn; ++t) {
    float* hin  = (t & 1) ? h1 : h0;
    float* hout = (t & 1) ? h0 : h1;
    lstm_step_kernel<<<128 / 4, 128, 0, stream>>>(zall, t, Uh, rmask, hin,
                                                  hout, cbuf, hsbuf);
  }

  // 5) logits: 66 m-groups * 125 n-groups = 8250 waves, 2 waves/block
  gemm_head_kernel<<<8250 / 2, 64, 0, stream>>>(hsbuf, fmask, Wd, bd, out);
}